// GraphSAGELayer_38774964748364
// MI455X (gfx1250) — compile-verified
//
#include <hip/hip_runtime.h>

typedef __attribute__((ext_vector_type(2))) float v2f;
typedef __attribute__((ext_vector_type(8))) float v8f;

#define SAGE_F   128
#define SAGE_TI  16
#define SAGE_NEG (-1e30f)
#define SAGE_EPS 1e-5f

__global__ __launch_bounds__(256)
void sage_fused_kernel(const float* __restrict__ h,
                       const int*   __restrict__ adj,
                       const float* __restrict__ Wself,
                       const float* __restrict__ bself,
                       const float* __restrict__ Wneigh,
                       const float* __restrict__ bneigh,
                       const float* __restrict__ gamma,
                       const float* __restrict__ beta,
                       float* __restrict__ out,
                       int N)
{
    extern __shared__ unsigned lds_mask[];            // N entries, 16-bit packed adjacency
    __shared__ float lds_h[SAGE_TI][SAGE_F];          // h tile
    __shared__ float lds_neigh[SAGE_TI][SAGE_F];      // max-aggregated neighbors
    __shared__ float lds_buf[SAGE_TI][SAGE_F];        // half-1 partial maxes, then output tile
    __shared__ unsigned lds_or[2];
    __shared__ float lds_mu[SAGE_TI];
    __shared__ float lds_rs[SAGE_TI];

    const int t    = (int)threadIdx.x;
    const int f    = t & (SAGE_F - 1);
    const int half = t >> 7;                          // 0 or 1
    const int lane = t & 31;
    const int wave = t >> 5;                          // 0..7

    const int tilesPerB = N / SAGE_TI;
    const int b  = (int)blockIdx.x / tilesPerB;
    const int i0 = ((int)blockIdx.x % tilesPerB) * SAGE_TI;

    const float* __restrict__ hb   = h   + (size_t)b * N * SAGE_F;
    const int*   __restrict__ adjb = adj + (size_t)b * N * N;

    // ---- Stage h tile into LDS ----
    for (int idx = t; idx < SAGE_TI * SAGE_F; idx += 256) {
        int r = idx >> 7, c = idx & (SAGE_F - 1);
        lds_h[r][c] = hb[(size_t)(i0 + r) * SAGE_F + c];
    }

    // ---- Build packed adjacency masks: bit ii of lds_mask[j] = edge(i0+ii -> j), no self-loops ----
    for (int j = t; j < N; j += 256) {
        unsigned m = 0u;
        #pragma unroll
        for (int ii = 0; ii < SAGE_TI; ++ii) {
            int node = i0 + ii;
            int a = adjb[(size_t)node * N + j];
            if (a != 0 && j != node) m |= (1u << ii);
        }
        lds_mask[j] = m;
    }
    __syncthreads();

    // ---- Max aggregation: j range split across the two block halves ----
    float acc[SAGE_TI];
    #pragma unroll
    for (int ii = 0; ii < SAGE_TI; ++ii) acc[ii] = SAGE_NEG;
    unsigned orAcc = 0u;
    {
        const int jb = half * (N >> 1), je = jb + (N >> 1);
        for (int j = jb; j < je; ++j) {
            unsigned m = lds_mask[j];
            orAcc |= m;
            float hv = hb[(size_t)j * SAGE_F + f];    // coalesced 128B per wave
            #pragma unroll
            for (int ii = 0; ii < SAGE_TI; ++ii) {
                float cand = (m & (1u << ii)) ? hv : SAGE_NEG;
                acc[ii] = fmaxf(acc[ii], cand);
            }
        }
    }
    if (half == 1) {
        #pragma unroll
        for (int ii = 0; ii < SAGE_TI; ++ii) lds_buf[ii][f] = acc[ii];
    }
    if (f == 0) lds_or[half] = orAcc;                 // orAcc uniform within each half
    __syncthreads();
    const unsigned orBoth = lds_or[0] | lds_or[1];
    if (half == 0) {
        #pragma unroll
        for (int ii = 0; ii < SAGE_TI; ++ii) {
            float mx = fmaxf(acc[ii], lds_buf[ii][f]);
            lds_neigh[ii][f] = ((orBoth >> ii) & 1u) ? mx : lds_h[ii][f];
        }
    }
    __syncthreads();

    // ---- GEMM: out[16,128] = lds_h@Wself + lds_neigh@Wneigh + biases ----
    // Wave w owns output column tile n0 = 16*w; V_WMMA_F32_16X16X4_F32, K-loop over 128.
    // A (16x4 f32): lanes 0-15 -> M, K={k,k+1}; lanes 16-31 -> M, K={k+2,k+3}
    // B (4x16 f32): VGPR0 rows k / k+2, VGPR1 rows k+1 / k+3, N = lane&15
    const int n0   = wave * 16;
    const int mrow = lane & 15;
    const int kb   = (lane >> 4) * 2;
    const int nn   = lane & 15;
    v8f c = {};
    #pragma unroll 4
    for (int k = 0; k < SAGE_F; k += 4) {
        v2f a, bb;
        a.x  = lds_h[mrow][k + kb];
        a.y  = lds_h[mrow][k + kb + 1];
        bb.x = Wself[(size_t)(k + kb)     * SAGE_F + n0 + nn];
        bb.y = Wself[(size_t)(k + kb + 1) * SAGE_F + n0 + nn];
        c = __builtin_amdgcn_wmma_f32_16x16x4_f32(false, a, false, bb, (short)0, c, false, false);
    }
    #pragma unroll 4
    for (int k = 0; k < SAGE_F; k += 4) {
        v2f a, bb;
        a.x  = lds_neigh[mrow][k + kb];
        a.y  = lds_neigh[mrow][k + kb + 1];
        bb.x = Wneigh[(size_t)(k + kb)     * SAGE_F + n0 + nn];
        bb.y = Wneigh[(size_t)(k + kb + 1) * SAGE_F + n0 + nn];
        c = __builtin_amdgcn_wmma_f32_16x16x4_f32(false, a, false, bb, (short)0, c, false, false);
    }
    const float bias = bself[n0 + nn] + bneigh[n0 + nn];
    #pragma unroll
    for (int r = 0; r < 8; ++r) {
        int mr = r + (lane >> 4) * 8;                 // C layout: VGPR r -> M=r / M=r+8
        lds_buf[mr][n0 + nn] = c[r] + bias;
    }
    __syncthreads();

    // ---- LayerNorm stats (population var, matching jnp.var) ----
    if (t < SAGE_TI) {
        float s = 0.f, s2 = 0.f;
        for (int c2 = 0; c2 < SAGE_F; ++c2) {
            float v = lds_buf[t][c2];
            s += v; s2 += v * v;
        }
        float mu  = s  * (1.0f / SAGE_F);
        float var = s2 * (1.0f / SAGE_F) - mu * mu;
        lds_mu[t] = mu;
        lds_rs[t] = rsqrtf(var + SAGE_EPS);
    }
    __syncthreads();

    // ---- Affine + ReLU, coalesced stores ----
    const float g = gamma[f], be = beta[f];
    #pragma unroll
    for (int r8 = 0; r8 < 8; ++r8) {
        int r = half * 8 + r8;
        float v = (lds_buf[r][f] - lds_mu[r]) * lds_rs[r] * g + be;
        out[((size_t)b * N + i0 + r) * SAGE_F + f] = fmaxf(v, 0.f);
    }
}

extern "C" void kernel_launch(void* const* d_in, const int* in_sizes, int n_in,
                              void* d_out, int out_size, void* d_ws, size_t ws_size,
                              hipStream_t stream) {
    (void)n_in; (void)out_size; (void)d_ws; (void)ws_size;
    const float* h      = (const float*)d_in[0];
    const int*   adj    = (const int*)  d_in[1];
    const float* Wself  = (const float*)d_in[2];
    const float* bself  = (const float*)d_in[3];
    const float* Wneigh = (const float*)d_in[4];
    const float* bneigh = (const float*)d_in[5];
    const float* gamma  = (const float*)d_in[6];
    const float* beta   = (const float*)d_in[7];
    float* out = (float*)d_out;

    const int BN = in_sizes[0] / SAGE_F;   // B*N
    const int N  = in_sizes[1] / BN;       // adj has B*N*N elements
    const int blocks = BN / SAGE_TI;
    const size_t smem = (size_t)N * sizeof(unsigned);
    sage_fused_kernel<<<blocks, 256, smem, stream>>>(h, adj, Wself, bself,
                                                     Wneigh, bneigh, gamma, beta,
                                                     out, N);
}